// DecoderLayer_40312563040281
// MI455X (gfx1250) — compile-verified
//
#include <hip/hip_runtime.h>
#include <cstddef>

typedef __attribute__((ext_vector_type(2))) float f32x2;
typedef __attribute__((ext_vector_type(8))) float f32x8;

#define NVOX   64000          // 40*40*40
#define NCOL   128000         // B * NVOX
#define NTILE  8000           // NCOL / 16

// ---------------------------------------------------------------------------
// Prep kernels: layout transforms (tiny, run once per launch)
// ---------------------------------------------------------------------------

// occ_up mask: mask[b,z,y,x] = x_occ[b, z/2, y/2, x/2]
__global__ void prep_mask(const int* __restrict__ occ, float* __restrict__ mask) {
  int idx = blockIdx.x * blockDim.x + threadIdx.x;
  if (idx >= NCOL) return;
  int b = idx / NVOX, s = idx - b * NVOX;
  int Z = s / 1600, Y = (s / 40) % 40, X = s % 40;
  mask[idx] = (float)occ[((b * 20 + (Z >> 1)) * 20 + (Y >> 1)) * 20 + (X >> 1)];
}

// x_feat NCDHW -> channel-last xT[pcol][32]
__global__ void prep_xT(const float* __restrict__ xf, float* __restrict__ xT) {
  int idx = blockIdx.x * blockDim.x + threadIdx.x;
  if (idx >= 16000 * 32) return;
  int pcol = idx >> 5, c = idx & 31;
  int b = pcol / 8000, ps = pcol - b * 8000;
  xT[idx] = xf[((size_t)(b * 32 + c)) * 8000 + ps];
}

// W_up[c][d][p][q][r] -> WupT[parity][d][c]
__global__ void prep_wup(const float* __restrict__ w, float* __restrict__ wT) {
  int idx = blockIdx.x * blockDim.x + threadIdx.x;
  if (idx >= 8 * 64 * 32) return;
  int parity = idx / (64 * 32);
  int d = (idx / 32) % 64;
  int c = idx & 31;
  wT[idx] = w[((size_t)(c * 64 + d)) * 8 + parity];
}

// W[o][i][taps] (o < Cout) -> Wt[o][t][i], o zero-padded to Mpad
__global__ void wtrans(const float* __restrict__ src, float* __restrict__ dst,
                       int Cout, int Cin, int taps, int Mpad) {
  int idx = blockIdx.x * blockDim.x + threadIdx.x;
  int tot = Mpad * taps * Cin;
  if (idx >= tot) return;
  int o = idx / (taps * Cin);
  int rem = idx - o * taps * Cin;
  int t = rem / Cin;
  int i = rem - t * Cin;
  dst[idx] = (o < Cout) ? src[((size_t)(o * Cin + i)) * taps + t] : 0.f;
}

// ---------------------------------------------------------------------------
// WMMA upsample: 8 per-parity GEMMs, M=64 (4 M-tiles in-wave, shared B frag),
// K=32, N = 16000 parent columns.  h[col][64] = relu((Wup x + b) * m)
// ---------------------------------------------------------------------------
__global__ void upsample_wmma(const float* __restrict__ xT, const float* __restrict__ wupT,
                              const float* __restrict__ bup, const float* __restrict__ mask,
                              float* __restrict__ h, int totalWaves) {
  int wid = blockIdx.x * (blockDim.x >> 5) + (threadIdx.x >> 5);
  if (wid >= totalWaves) return;
  int lane = threadIdx.x & 31;
  int n = lane & 15, hi = lane >> 4;

  int parity = wid & 7;
  int tile = wid >> 3;

  int pcol = tile * 16 + n;                 // [0, 16000)
  int b = pcol / 8000, ps = pcol - b * 8000;
  int PZ = ps / 400, PY = (ps / 20) % 20, PX = ps % 20;
  int p = (parity >> 2) & 1, q = (parity >> 1) & 1, r = parity & 1;
  int Z = 2 * PZ + p, Y = 2 * PY + q, X = 2 * PX + r;
  int col = b * NVOX + (Z * 40 + Y) * 40 + X;

  const float* bp = xT + (size_t)pcol * 32 + hi * 2;
  const float* ap[4];
  #pragma unroll
  for (int m = 0; m < 4; ++m)
    ap[m] = wupT + ((size_t)parity * 64 + m * 16 + n) * 32 + hi * 2;

  f32x8 acc[4] = {};
  #pragma unroll
  for (int c = 0; c < 32; c += 4) {
    f32x2 bf = *(const f32x2*)(bp + c);
    #pragma unroll
    for (int m = 0; m < 4; ++m) {
      f32x2 af = *(const f32x2*)(ap[m] + c);
      acc[m] = __builtin_amdgcn_wmma_f32_16x16x4_f32(false, af, false, bf, (short)0,
                                                     acc[m], false, false);
    }
  }

  float mv = mask[col];
  #pragma unroll
  for (int m = 0; m < 4; ++m) {
    #pragma unroll
    for (int j = 0; j < 8; ++j) {
      int ch = m * 16 + j + hi * 8;
      float v = (acc[m][j] + bup[ch]) * mv;
      v = fmaxf(v, 0.f);
      h[(size_t)col * 64 + ch] = v;
    }
  }
}

// ---------------------------------------------------------------------------
// Implicit-GEMM conv (SAME, KS=1 or 3), compile-time CIN/KS/MT so the K-loop
// fully unrolls per tap and MT accumulator chains share each B fragment.
// Fused epilogue: (acc + bias) * mask (*relu) (+resid into concat slot).
// in: [col][CIN] channel-last.  wt: [MT*16][taps*CIN] zero-padded rows.
// ---------------------------------------------------------------------------
template <int CIN, int KS, int MT>
__global__ void conv3d_wmma(const float* __restrict__ in, const float* __restrict__ wt,
                            const float* __restrict__ bias, const float* __restrict__ mask,
                            const float* __restrict__ resid, float* __restrict__ outp,
                            int CoutActual, int Cstore, int coutBase, int doRelu,
                            int totalWaves) {
  int wid = blockIdx.x * (blockDim.x >> 5) + (threadIdx.x >> 5);
  if (wid >= totalWaves) return;
  int lane = threadIdx.x & 31;
  int n = lane & 15, hi = lane >> 4;
  int tile = wid;

  int col = tile * 16 + n;                  // exact: NCOL % 16 == 0
  int b = col / NVOX, s = col - b * NVOX;
  int Z = s / 1600, Y = (s / 40) % 40, X = s % 40;

  constexpr int NT = KS * KS * KS;
  constexpr int KTOT = NT * CIN;
  const float* ap0[MT];
  #pragma unroll
  for (int m = 0; m < MT; ++m)
    ap0[m] = wt + ((size_t)(m * 16 + n)) * KTOT + hi * 2;

  f32x8 acc[MT] = {};
  const f32x2 bzero = {0.f, 0.f};
  #pragma unroll 1
  for (int t = 0; t < NT; ++t) {
    int dz = 0, dy = 0, dx = 0;
    if (KS == 3) { dz = t / 9 - 1; dy = (t / 3) % 3 - 1; dx = t % 3 - 1; }
    int Zs = Z + dz, Ys = Y + dy, Xs = X + dx;
    bool valid = ((unsigned)Zs < 40u) && ((unsigned)Ys < 40u) && ((unsigned)Xs < 40u);
    const float* bp = in + (size_t)(b * NVOX + (Zs * 40 + Ys) * 40 + Xs) * CIN + hi * 2;
    #pragma unroll
    for (int c = 0; c < CIN; c += 4) {
      f32x2 bf = valid ? *(const f32x2*)(bp + c) : bzero;
      #pragma unroll
      for (int m = 0; m < MT; ++m) {
        f32x2 af = *(const f32x2*)(ap0[m] + t * CIN + c);
        acc[m] = __builtin_amdgcn_wmma_f32_16x16x4_f32(false, af, false, bf, (short)0,
                                                       acc[m], false, false);
      }
    }
  }

  float mv = mask[col];
  #pragma unroll
  for (int m = 0; m < MT; ++m) {
    #pragma unroll
    for (int j = 0; j < 8; ++j) {
      int c = m * 16 + j + hi * 8;
      if (c < CoutActual) {
        float v = (acc[m][j] + bias[c]) * mv;
        if (doRelu) v = fmaxf(v, 0.f);
        size_t oidx = (size_t)col * Cstore + coutBase + c;
        if (resid) v += resid[oidx];
        outp[oidx] = v;
      }
    }
  }
}

// ---------------------------------------------------------------------------
// keep-mask machinery (adaptive top-k, exact stable-rank semantics)
// ---------------------------------------------------------------------------
__global__ void counts_kernel(const float* __restrict__ mask, const int* __restrict__ tgt,
                              int* __restrict__ counts) {
  int idx = blockIdx.x * blockDim.x + threadIdx.x;
  if (idx >= NCOL) return;
  int b = idx / NVOX;
  if (mask[idx] > 0.f) atomicAdd(&counts[b], 1);
  if (tgt[idx] != 0)   atomicAdd(&counts[2 + b], 1);
}

// rank_i = #{j: v_j > v_i} + #{j: v_j == v_i, j < i}  (stable descending rank)
__global__ void rank_kernel(const float* __restrict__ cls, const float* __restrict__ mask,
                            int* __restrict__ rank) {
  int b = blockIdx.x / 250;
  int iL = (blockIdx.x % 250) * 256 + threadIdx.x;   // 64000 = 250*256 exact
  int i = b * NVOX + iL;
  float ninf = -__builtin_inff();
  float myv = (mask[i] > 0.f) ? cls[i] : ninf;
  __shared__ float sv[256];
  int cnt = 0;
  for (int base = 0; base < NVOX; base += 256) {
    int j = b * NVOX + base + threadIdx.x;
    sv[threadIdx.x] = (mask[j] > 0.f) ? cls[j] : ninf;
    __syncthreads();
    for (int t = 0; t < 256; ++t) {
      float vj = sv[t];
      int jg = base + t;
      cnt += ((vj > myv) || (vj == myv && jg < iL)) ? 1 : 0;
    }
    __syncthreads();
  }
  rank[i] = cnt;
}

__global__ void keep_kernel(const float* __restrict__ mask, const int* __restrict__ tgt,
                            const int* __restrict__ rank, const float* __restrict__ cls,
                            const int* __restrict__ counts, const int* __restrict__ adaptive,
                            float* __restrict__ keepf, float* __restrict__ targf) {
  int idx = blockIdx.x * blockDim.x + threadIdx.x;
  if (idx >= NCOL) return;
  int b = idx / NVOX;
  bool occ = mask[idx] > 0.f;
  bool tg = occ && (tgt[idx] != 0);
  bool kp;
  if (*adaptive) {
    int k = min(counts[b], counts[2 + b]);   // rho == 1.0
    kp = occ && (rank[idx] < k);
  } else {
    kp = occ && (cls[idx] > 0.f);
  }
  kp = kp || tg;
  keepf[idx] = kp ? 1.f : 0.f;
  targf[idx] = tg ? 1.f : 0.f;
}

// outputs: out_pruned (NCDHW), out_cls, target, keep  -> flat f32
__global__ void final_kernel(const float* __restrict__ outF, const float* __restrict__ cls,
                             const float* __restrict__ keepf, const float* __restrict__ targf,
                             float* __restrict__ dout) {
  int idx = blockIdx.x * blockDim.x + threadIdx.x;
  if (idx >= 4480000) return;
  if (idx < 4096000) {
    int b = idx / (32 * NVOX);
    int c = (idx / NVOX) % 32;
    int s = idx % NVOX;
    int col = b * NVOX + s;
    dout[idx] = outF[(size_t)col * 32 + c] * keepf[col];
  } else if (idx < 4224000) {
    dout[idx] = cls[idx - 4096000];
  } else if (idx < 4352000) {
    dout[idx] = targf[idx - 4224000];
  } else {
    dout[idx] = keepf[idx - 4352000];
  }
}

// ---------------------------------------------------------------------------
extern "C" void kernel_launch(void* const* d_in, const int* in_sizes, int n_in,
                              void* d_out, int out_size, void* d_ws, size_t ws_size,
                              hipStream_t stream) {
  (void)in_sizes; (void)n_in; (void)out_size; (void)ws_size;
  const float* x_feat   = (const float*)d_in[0];
  const int*   x_occ    = (const int*)d_in[1];
  const int*   tgt_occ  = (const int*)d_in[2];
  const float* W_up     = (const float*)d_in[3];
  const float* b_up     = (const float*)d_in[4];
  const float* W_conv   = (const float*)d_in[5];
  const float* b_conv   = (const float*)d_in[6];
  const float* bW00     = (const float*)d_in[7];
  const float* bb00     = (const float*)d_in[8];
  const float* bW01     = (const float*)d_in[9];
  const float* bb01     = (const float*)d_in[10];
  const float* bW10     = (const float*)d_in[11];
  const float* bb10     = (const float*)d_in[12];
  const float* bW11     = (const float*)d_in[13];
  const float* bb11     = (const float*)d_in[14];
  const float* bW12     = (const float*)d_in[15];
  const float* bb12     = (const float*)d_in[16];
  const float* W_cls    = (const float*)d_in[17];
  const float* b_cls    = (const float*)d_in[18];
  const int*   adaptive = (const int*)d_in[19];

  float* ws = (float*)d_ws;
  size_t off = 0;
  auto take = [&](size_t nf) { float* p = ws + off; off += (nf + 3) & ~(size_t)3; return p; };

  float* xT    = take(512000);
  float* h     = take((size_t)NCOL * 64);
  float* outA  = take((size_t)NCOL * 32);
  float* outB  = take((size_t)NCOL * 32);
  float* t8a   = take((size_t)NCOL * 8);
  float* t8b   = take((size_t)NCOL * 8);
  float* t8c   = take((size_t)NCOL * 8);
  float* maskb = take(NCOL);
  float* clsb  = take(NCOL);
  float* keepf = take(NCOL);
  float* targf = take(NCOL);
  int*   rankb = (int*)take(NCOL);
  int*   cnts  = (int*)take(8);
  float* WupT  = take(8 * 64 * 32);
  float* WconvT= take(32 * 27 * 64);
  float* WclsT = take(16 * 27 * 32);
  float* W00T[3]; float* W01T[3]; float* W10T[3]; float* W11T[3]; float* W12T[3];
  for (int i = 0; i < 3; ++i) {
    W00T[i] = take(16 * 27 * 32);
    W01T[i] = take(16 * 27 * 8);
    W10T[i] = take(16 * 1 * 32);
    W11T[i] = take(16 * 27 * 8);
    W12T[i] = take(16 * 1 * 8);
  }

  hipMemsetAsync(cnts, 0, 4 * sizeof(int), stream);

  prep_mask<<<(NCOL + 255) / 256, 256, 0, stream>>>(x_occ, maskb);
  prep_xT<<<(512000 + 255) / 256, 256, 0, stream>>>(x_feat, xT);
  prep_wup<<<(16384 + 255) / 256, 256, 0, stream>>>(W_up, WupT);
  wtrans<<<(55296 + 255) / 256, 256, 0, stream>>>(W_conv, WconvT, 32, 64, 27, 32);
  wtrans<<<(13824 + 255) / 256, 256, 0, stream>>>(W_cls, WclsT, 1, 32, 27, 16);
  for (int i = 0; i < 3; ++i) {
    wtrans<<<(13824 + 255) / 256, 256, 0, stream>>>(bW00 + (size_t)i * 8 * 32 * 27, W00T[i], 8, 32, 27, 16);
    wtrans<<<(3456  + 255) / 256, 256, 0, stream>>>(bW01 + (size_t)i * 16 * 8 * 27, W01T[i], 16, 8, 27, 16);
    wtrans<<<(512   + 255) / 256, 256, 0, stream>>>(bW10 + (size_t)i * 8 * 32,      W10T[i], 8, 32, 1, 16);
    wtrans<<<(3456  + 255) / 256, 256, 0, stream>>>(bW11 + (size_t)i * 8 * 8 * 27,  W11T[i], 8, 8, 27, 16);
    wtrans<<<(128   + 255) / 256, 256, 0, stream>>>(bW12 + (size_t)i * 16 * 8,      W12T[i], 16, 8, 1, 16);
  }

  // Upsample: 1000 tiles * 8 parities = 8000 waves (4 M-tiles in-wave)
  upsample_wmma<<<1000, 256, 0, stream>>>(xT, WupT, b_up, maskb, h, 8000);

  // conv 64->32 + relu (2 M-tiles in-wave, shared B)
  conv3d_wmma<64, 3, 2><<<1000, 256, 0, stream>>>(h, WconvT, b_conv, maskb, nullptr, outA,
                                                  32, 32, 0, 1, NTILE);

  float* cur = outA;
  float* nxt = outB;
  for (int i = 0; i < 3; ++i) {
    // branch 0: 3x3x3 32->8 relu ; 3x3x3 8->16 (+residual into concat[0:16))
    conv3d_wmma<32, 3, 1><<<1000, 256, 0, stream>>>(cur, W00T[i], bb00 + i * 8, maskb,
                                                    nullptr, t8a, 8, 8, 0, 1, NTILE);
    conv3d_wmma<8, 3, 1><<<1000, 256, 0, stream>>>(t8a, W01T[i], bb01 + i * 16, maskb,
                                                   cur, nxt, 16, 32, 0, 0, NTILE);
    // branch 1: 1x1x1 32->8 relu ; 3x3x3 8->8 relu ; 1x1x1 8->16 (+res into [16:32))
    conv3d_wmma<32, 1, 1><<<1000, 256, 0, stream>>>(cur, W10T[i], bb10 + i * 8, maskb,
                                                    nullptr, t8b, 8, 8, 0, 1, NTILE);
    conv3d_wmma<8, 3, 1><<<1000, 256, 0, stream>>>(t8b, W11T[i], bb11 + i * 8, maskb,
                                                   nullptr, t8c, 8, 8, 0, 1, NTILE);
    conv3d_wmma<8, 1, 1><<<1000, 256, 0, stream>>>(t8c, W12T[i], bb12 + i * 16, maskb,
                                                   cur, nxt, 16, 32, 16, 0, NTILE);
    float* tmp = cur; cur = nxt; nxt = tmp;
  }

  // classifier conv 32->1
  conv3d_wmma<32, 3, 1><<<1000, 256, 0, stream>>>(cur, WclsT, b_cls, maskb, nullptr, clsb,
                                                  1, 1, 0, 0, NTILE);

  counts_kernel<<<(NCOL + 255) / 256, 256, 0, stream>>>(maskb, tgt_occ, cnts);
  rank_kernel<<<500, 256, 0, stream>>>(clsb, maskb, rankb);
  keep_kernel<<<(NCOL + 255) / 256, 256, 0, stream>>>(maskb, tgt_occ, rankb, clsb,
                                                      cnts, adaptive, keepf, targf);
  final_kernel<<<(4480000 + 255) / 256, 256, 0, stream>>>(cur, clsb, keepf, targf,
                                                          (float*)d_out);
}